// Net_19086834664186
// MI455X (gfx1250) — compile-verified
//
#include <hip/hip_runtime.h>
#include <hip/hip_bf16.h>

typedef __attribute__((ext_vector_type(2))) float v2f;
typedef __attribute__((ext_vector_type(8))) float v8f;

#define NBASIS 25
#define TSTEPS 300
#define NCOLS 602          // 2 * 301
#define NCOLS_PAD 608      // 38 * 16
#define KDIM 55
#define KPAD 56            // 14 * 4
#define COEF_STRIDE 27     // A_t, G_t, Phi_t[25]

// ---------------------------------------------------------------------------
// Kernel 1: 27-lane affine-coefficient recursion for the DMP scan.
// traj[b,d,t] = A_t*y0 + G_t*g + sum_n Phi_t[n]*w[b,d,n]  (per dof, dofs identical)
// Lane 0 -> A, lane 1 -> G, lanes 2..26 -> Phi[n].
// ---------------------------------------------------------------------------
__global__ __launch_bounds__(32) void dmp_coef_kernel(float* __restrict__ coef) {
    __shared__ float c[NBASIS], s2[NBASIS];
    const int lane = threadIdx.x;
    if (lane < NBASIS) c[lane] = expf(-2.0f * (float)lane / 24.0f);
    __syncthreads();
    if (lane < NBASIS) {
        int j = (lane < NBASIS - 1) ? lane : NBASIS - 2;
        float d = (c[j + 1] - c[j]) * 0.75f;
        s2[lane] = d * d;
    }
    __syncthreads();

    const float hstep = 0.01f / 3.0f;     // DT / TAU
    const float AZ = 48.0f, BZ = 12.0f, AX = 2.0f;

    float Y = (lane == 0) ? 1.0f : 0.0f;  // coefficient on y0
    float Z = 0.0f;
    if (lane < COEF_STRIDE) coef[lane] = Y;  // t = 0 row (traj[:,:,0] = y0)

    float xp = 1.0f;
    const float eg = (lane == 1) ? 1.0f : 0.0f;  // coefficient source on g
    for (int t = 1; t <= TSTEPS; ++t) {
        float ssum = 0.0f, mypsi = 0.0f;
        for (int n = 0; n < NBASIS; ++n) {
            float dd = xp - c[n];
            float p = expf(-0.5f * dd * dd / s2[n]);
            ssum += p;
            if (n == lane - 2) mypsi = p;
        }
        float phi = (lane >= 2 && lane < COEF_STRIDE) ? mypsi * xp / ssum : 0.0f;
        float dz = (AZ * (BZ * (eg - Y) - Z) + phi) * hstep;
        float dy = Z * hstep;
        Y += dy;
        Z += dz;
        xp -= AX * xp * hstep;
        if (lane < COEF_STRIDE) coef[t * COEF_STRIDE + lane] = Y;
    }
}

// ---------------------------------------------------------------------------
// Kernel 2: fold W / b through the coefficient table.
// Q[k, col] = sum_j W[j,k] * C[j,col]   (C sparse: rows 1+d, 3+d, 5+25d+n)
// Zero-padded to 56 x 608; row index 56 of the grid computes r[col] from b.
// ---------------------------------------------------------------------------
__global__ void build_Q_kernel(const float* __restrict__ W,
                               const float* __restrict__ bvec,
                               const float* __restrict__ coef,
                               float* __restrict__ Q,
                               float* __restrict__ r) {
    int col = blockIdx.x * blockDim.x + threadIdx.x;  // 0..607
    int row = blockIdx.y;                             // 0..56 (56 == r row)
    if (col >= NCOLS_PAD) return;

    float acc = 0.0f;
    if (col < NCOLS) {
        int d = col / 301;
        const float* cf = coef + (col % 301) * COEF_STRIDE;
        if (row == KPAD) {  // r from bias
            acc = bvec[1 + d] * cf[0] + bvec[3 + d] * cf[1];
            for (int n = 0; n < NBASIS; ++n)
                acc += bvec[5 + NBASIS * d + n] * cf[2 + n];
        } else if (row < KDIM) {
            acc = W[(1 + d) * KDIM + row] * cf[0] + W[(3 + d) * KDIM + row] * cf[1];
            for (int n = 0; n < NBASIS; ++n)
                acc += W[(5 + NBASIS * d + n) * KDIM + row] * cf[2 + n];
        }
    }
    if (row == KPAD) r[col] = acc;
    else             Q[row * NCOLS_PAD + col] = acc;
}

// ---------------------------------------------------------------------------
// Kernel 3: out(65536 x 602) = x(65536 x 55) @ Q(55 x 602) + r, via
// V_WMMA_F32_16X16X4_F32.  One block = 16 batch rows, 8 waves sweep 38
// N-tiles of 16 columns.  A/B fragments per ISA 32-bit layouts:
//   A 16x4 : lanes 0-15 hold (K0,K1), lanes 16-31 hold (K2,K3), M = lane&15
//   B 4x16 : lanes 0-15 hold (K0,K1), lanes 16-31 hold (K2,K3), N = lane&15
//   C/D    : VGPR v -> M = v (+8 for upper half-wave), N = lane&15
// ---------------------------------------------------------------------------
__global__ __launch_bounds__(256) void gemm_kernel(const float* __restrict__ x,
                                                   const float* __restrict__ Q,
                                                   const float* __restrict__ r,
                                                   float* __restrict__ out) {
    __shared__ float xs[16 * 57];  // 16 rows x KPAD, stride 57 vs bank conflicts
    const int tid = threadIdx.x;
    const int rowbase = blockIdx.x * 16;

    for (int i = tid; i < 16 * KPAD; i += 256) {
        int rr = i / KPAD, kk = i % KPAD;
        xs[rr * 57 + kk] = (kk < KDIM) ? x[(size_t)(rowbase + rr) * KDIM + kk] : 0.0f;
    }
    __syncthreads();

    const int wave = tid >> 5;
    const int lane = tid & 31;
    const int m = lane & 15;             // M (for A) / N (for B) within tile
    const int khalf = (lane >> 4) * 2;   // 0 for lanes 0-15, 2 for lanes 16-31

    for (int tile = wave; tile < NCOLS_PAD / 16; tile += 8) {
        const int col = tile * 16 + m;
        v8f acc = {};
#pragma unroll
        for (int k0 = 0; k0 < KPAD; k0 += 4) {
            v2f afr, bfr;
            afr.x = xs[m * 57 + k0 + khalf];
            afr.y = xs[m * 57 + k0 + khalf + 1];
            bfr.x = Q[(k0 + khalf) * NCOLS_PAD + col];
            bfr.y = Q[(k0 + khalf + 1) * NCOLS_PAD + col];
            acc = __builtin_amdgcn_wmma_f32_16x16x4_f32(
                false, afr, false, bfr, (short)0, acc, false, false);
        }
        if (col < NCOLS) {
            const float rv = r[col];
            const int mbase = rowbase + ((lane >> 4) ? 8 : 0);
#pragma unroll
            for (int v = 0; v < 8; ++v) {
                out[(size_t)(mbase + v) * NCOLS + col] = acc[v] + rv;
            }
        }
    }
}

// ---------------------------------------------------------------------------
extern "C" void kernel_launch(void* const* d_in, const int* in_sizes, int n_in,
                              void* d_out, int out_size, void* d_ws, size_t ws_size,
                              hipStream_t stream) {
    const float* x = (const float*)d_in[0];   // 65536 x 55
    const float* W = (const float*)d_in[1];   // 55 x 55
    const float* b = (const float*)d_in[2];   // 55
    float* out = (float*)d_out;               // 65536 x 602

    float* ws = (float*)d_ws;
    float* coef = ws;                       // 301*27 = 8127 floats (rounded 8192)
    float* Q    = ws + 8192;                // 56*608 = 34048 floats
    float* r    = ws + 8192 + (KPAD * NCOLS_PAD);  // 608 floats

    dmp_coef_kernel<<<1, 32, 0, stream>>>(coef);

    dim3 g2((NCOLS_PAD + 63) / 64, KPAD + 1);
    build_Q_kernel<<<g2, 64, 0, stream>>>(W, b, coef, Q, r);

    gemm_kernel<<<65536 / 16, 256, 0, stream>>>(x, Q, r, out);
}